// _GridExtractor_23201413333176
// MI455X (gfx1250) — compile-verified
//
#include <hip/hip_runtime.h>
#include <hip/hip_bf16.h>

typedef float v2f __attribute__((ext_vector_type(2)));
typedef float v8f __attribute__((ext_vector_type(8)));

#define NLVL   256
#define IMG_H  96
#define IMG_W  96
#define NPIX   (IMG_H * IMG_W)
#define NBATCH 2
#define TOTPIX (NBATCH * NPIX)

// ---------------------------------------------------------------------------
// Kernel 1: per-pixel level moments via V_WMMA_F32_16X16X4_F32.
// Each wave handles 16 pixels (M rows). K loop over 256 levels in steps of 4.
//   A[M,K]   = exp(-2*(x_M - c_K)^2)          (soft bin value, f32)
//   B[K,N]   = [1, K, K^2, 0...] for N=0..2   (branch-free: w0 + l*(w1 + w2*l))
//   C[M,N]   accumulates m0,m1,m2 per pixel in columns 0..2.
// Layouts per CDNA5 ISA 7.12.2 (32-bit 16x4 A, 4x16 B, 16x16 C):
//   lanes 0-15 hold K pair {0,1}, lanes 16-31 hold K pair {2,3};
//   C: VGPR r -> (M=r, N=lane) for lanes 0-15, (M=8+r, N=lane-16) otherwise.
// exp argument is in [-8,0] -> native v_exp (no range fixup needed).
// mom layout: 4 floats per pixel: [m0, m1, m2, e] where e = exp(-2*(x+1)^2)
// serves the linspace(-1,-1,256) quirk features (all centers == -1).
// ---------------------------------------------------------------------------
__global__ __launch_bounds__(256) void glcm_moments_wmma(const float* __restrict__ x,
                                                         float* __restrict__ mom) {
  const int lane = threadIdx.x & 31;
  const int wave = threadIdx.x >> 5;
  const int m    = lane & 15;   // M row (pixel) for A; N column for B/C
  const int half = lane >> 4;   // selects K pair {0,1} vs {2,3}
  const int base = (blockIdx.x * 8 + wave) * 16;   // TOTPIX = 144*8*16 exactly
  const float xv = x[base + m];

  // Per-lane moment-weight selectors (hoisted; no branches in the loop).
  const int n = m;
  const float w0 = (n == 0) ? 1.0f : 0.0f;
  const float w1 = (n == 1) ? 1.0f : 0.0f;
  const float w2 = (n == 2) ? 1.0f : 0.0f;

  v8f acc = {};
  float fl0 = (float)(2 * half);          // level index carried as float
  const float step = 2.0f / 255.0f;

#pragma unroll 4
  for (int j = 0; j < NLVL; j += 4) {
    const float fl1 = fl0 + 1.0f;
    const float d0 = xv - fmaf(fl0, step, -1.0f);
    const float d1 = xv - fmaf(fl1, step, -1.0f);
    v2f a, bm;
    a.x = __expf(-2.0f * d0 * d0);
    a.y = __expf(-2.0f * d1 * d1);
    bm.x = w0 + fl0 * (w1 + w2 * fl0);
    bm.y = w0 + fl1 * (w1 + w2 * fl1);
    // 8 args: (neg_a, A, neg_b, B, c_mod, C, reuse_a, reuse_b)
    acc = __builtin_amdgcn_wmma_f32_16x16x4_f32(false, a, false, bm, (short)0, acc,
                                                false, false);
    fl0 += 4.0f;
  }

  // Scatter columns 0..2 of C (m0,m1,m2 for the wave's 16 pixels).
  if (n < 3) {
#pragma unroll
    for (int r = 0; r < 8; ++r) {
      const int M = r + 8 * half;
      mom[(size_t)(base + M) * 4 + n] = acc[r];
    }
  }
  // Degenerate-center exponential for theta in {90,135} (centers all -1).
  if (half == 0) {
    const float dd = xv + 1.0f;
    mom[(size_t)(base + m) * 4 + 3] = __expf(-2.0f * dd * dd);
  }
}

// ---------------------------------------------------------------------------
// Kernel 2: one block per output scalar (16 features x 2 batches = 32).
// contrast = N / D with
//   N = sum_valid  m2(p)*m0(q) - 2*m1(p)*m1(q) + m0(p)*m2(q)
//   D = sum_valid  m0(p)*m0(q)
// valid region / q per theta mirrors the reference roll+zero semantics.
// Reference guard: s = 2D; where s==0 -> 1, giving contrast = 2N.
// Output index: flat = feat*2 + batch (matches concat(axis=0).reshape(2,1,4,4)).
// ---------------------------------------------------------------------------
__global__ __launch_bounds__(256) void glcm_contrast(const float* __restrict__ mom,
                                                     float* __restrict__ out) {
  const int fb = blockIdx.x;      // 0..31
  const int f  = fb >> 1;         // feature 0..15
  const int b  = fb & 1;          // batch
  const int i  = f & 3;
  const int grp = f >> 2;         // 0:0deg 1:45 2:90 3:135
  const int d  = (grp & 1) ? ((i == 0) ? 1 : 2 * i)   // {1,2,4,6}
                           : (1 + 2 * i);             // {1,3,5,7}
  const bool setB = (f >= 8);     // linspace(-1,-1) features

  float N = 0.0f, D = 0.0f;
  for (int idx = threadIdx.x; idx < NPIX; idx += 256) {
    const int h = idx / IMG_W;
    const int w = idx - h * IMG_W;
    int qh, qw;
    bool valid;
    if (grp == 0)      { valid = (w < IMG_W - d);               qh = h;     qw = w + d; }
    else if (grp == 1) { valid = (h >= d) && (w < IMG_W - d);   qh = h - d; qw = w + d; }
    else if (grp == 2) { valid = (h >= d);                      qh = h - d; qw = w;     }
    else               { valid = (h >= d) && (w >= d);          qh = h - d; qw = w - d; }
    if (!valid) continue;

    const size_t p = (size_t)(b * NPIX + idx) * 4;
    const size_t q = (size_t)(b * NPIX + qh * IMG_W + qw) * 4;
    float m0p, m1p, m2p, m0q, m1q, m2q;
    if (!setB) {
      m0p = mom[p + 0]; m1p = mom[p + 1]; m2p = mom[p + 2];
      m0q = mom[q + 0]; m1q = mom[q + 1]; m2q = mom[q + 2];
    } else {
      const float ep = mom[p + 3], eq = mom[q + 3];
      m0p = 256.0f * ep; m1p = 32640.0f * ep; m2p = 5559680.0f * ep;
      m0q = 256.0f * eq; m1q = 32640.0f * eq; m2q = 5559680.0f * eq;
    }
    N += m2p * m0q - 2.0f * m1p * m1q + m0p * m2q;
    D += m0p * m0q;
  }

  __shared__ float sN[256];
  __shared__ float sD[256];
  sN[threadIdx.x] = N;
  sD[threadIdx.x] = D;
  __syncthreads();
  for (int s = 128; s > 0; s >>= 1) {
    if (threadIdx.x < s) {
      sN[threadIdx.x] += sN[threadIdx.x + s];
      sD[threadIdx.x] += sD[threadIdx.x + s];
    }
    __syncthreads();
  }
  if (threadIdx.x == 0) {
    const float num = sN[0], den = sD[0];
    out[fb] = (den == 0.0f) ? (2.0f * num) : (num / den);
  }
}

extern "C" void kernel_launch(void* const* d_in, const int* in_sizes, int n_in,
                              void* d_out, int out_size, void* d_ws, size_t ws_size,
                              hipStream_t stream) {
  (void)in_sizes; (void)n_in; (void)out_size; (void)ws_size;
  const float* x = (const float*)d_in[0];   // (2,1,96,96) f32
  float* mom = (float*)d_ws;                // TOTPIX * 4 floats = 288 KiB
  float* out = (float*)d_out;               // 32 f32

  glcm_moments_wmma<<<TOTPIX / (8 * 16), 256, 0, stream>>>(x, mom);
  glcm_contrast<<<32, 256, 0, stream>>>(mom, out);
}